// OneHotExclusivePooling_34857954574532
// MI455X (gfx1250) — compile-verified
//
#include <hip/hip_runtime.h>
#include <hip/hip_bf16.h>

// OneHotExclusivePooling for MI455X (gfx1250).
// Memory-bound scatter-add: zero 256MB output, then 4M events each add
// exp(-softplus(rate[ch])*dt) (M=2 adjacent floats) at seg*1024 + id*2.
//
// Roofline: ~256MB zero-store + 48MB event streams + ~4M random 64B-line
// atomic RMWs (mostly L2-resident; output 256MB vs 192MB L2) => ~20-35us
// floor at 23.3 TB/s. Compute (8M v_exp_f32) is negligible. No matrix
// structure -> WMMA inapplicable; the win is wide VMEM + L2 FP atomics.

#define FILTERS 128
#define SFM 1024        // STRIDE * FILTERS * CHANNEL_MULTIPLIER
#define TABLE (2 * FILTERS)

typedef float v4f __attribute__((ext_vector_type(4)));
typedef int   v4i __attribute__((ext_vector_type(4)));

// Non-returning FP32 atomic pair to adjacent floats: one address, two
// global_atomic_add_f32 (second via offset:4). No return -> STOREcnt only,
// wave never stalls on them (implicit wait at s_endpgm).
__device__ __forceinline__ void atomic_fadd_pair(float* p, float v0, float v1) {
    asm volatile(
        "global_atomic_add_f32 %0, %1, off\n\t"
        "global_atomic_add_f32 %0, %2, off offset:4"
        :: "v"(p), "v"(v0), "v"(v1)
        : "memory");
}

// ---------------------------------------------------------------------------
// Kernel 1: zero the output with wide B128 stores (grid-stride, 32-bit index,
// unrolled x4). Leaves the region dirty in the 192MB L2 so the subsequent
// atomics mostly resolve in L2.
// ---------------------------------------------------------------------------
__global__ void __launch_bounds__(256)
OneHot_zero_kernel(float* __restrict__ out, int n) {
    const int n4 = n >> 2;                 // 16,777,216 for the real shape
    v4f z;
    z.x = 0.0f; z.y = 0.0f; z.z = 0.0f; z.w = 0.0f;
    v4f* __restrict__ o4 = (v4f*)out;

    const int stride = (int)(gridDim.x * blockDim.x);
    int i = (int)(blockIdx.x * blockDim.x + threadIdx.x);

    // unrolled main loop: 4 independent b128 stores in flight per iteration
    for (; i + 3 * stride < n4; i += 4 * stride) {
        o4[(size_t)i]              = z;
        o4[(size_t)(i + stride)]   = z;
        o4[(size_t)(i + 2*stride)] = z;
        o4[(size_t)(i + 3*stride)] = z;
    }
    for (; i < n4; i += stride) {
        o4[(size_t)i] = z;
    }
    // generic scalar tail (n is divisible by 4 for the real shape)
    int tail = n & 3;
    if (blockIdx.x == 0 && (int)threadIdx.x < tail) {
        out[n - 1 - (int)threadIdx.x] = 0.0f;
    }
}

// ---------------------------------------------------------------------------
// Kernel 2: event scatter. Per block: build softplus(decay_rate) table (256
// floats) in LDS once (amortized over 1024 events), then each thread handles
// 4 consecutive events with one global_load_b128 per input stream,
// 2x v_exp_f32 and one fused pair of global_atomic_add_f32.
// ---------------------------------------------------------------------------
__global__ void __launch_bounds__(256)
OneHot_scatter_kernel(const float* __restrict__ dt,
                      const float* __restrict__ decay_rate,
                      const int*   __restrict__ ids,
                      const int*   __restrict__ segs,
                      float*       __restrict__ out,
                      int n_events) {
    __shared__ float rate[TABLE];

    const int t = threadIdx.x;
    if (t < TABLE) {
        // numerically stable softplus: max(x,0) + log1p(exp(-|x|))
        float x = decay_rate[t];
        rate[t] = fmaxf(x, 0.0f) + log1pf(__expf(-fabsf(x)));
    }
    __syncthreads();

    long long e = ((long long)blockIdx.x * blockDim.x + t) * 4;

    if (e + 3 < (long long)n_events) {
        // coalesced 16B-per-lane loads of all three event streams
        v4f d  = *(const v4f*)(dt   + e);
        v4i id = *(const v4i*)(ids  + e);
        v4i sg = *(const v4i*)(segs + e);
#pragma unroll
        for (int k = 0; k < 4; ++k) {
            int idk = id[k];
            int ch  = idk & (FILTERS - 1);          // id % 128 (F power of two)
            float r0 = rate[2 * ch + 0];            // ds_load (b64-mergeable)
            float r1 = rate[2 * ch + 1];
            float dk = d[k];
            float v0 = __expf(-r0 * dk);            // v_exp_f32
            float v1 = __expf(-r1 * dk);
            size_t o = (size_t)(unsigned)sg[k] * SFM + (size_t)(unsigned)idk * 2;
            atomic_fadd_pair(out + o, v0, v1);      // 2x global_atomic_add_f32
        }
    } else {
        // scalar tail
        for (; e < (long long)n_events; ++e) {
            int idk = ids[e];
            int ch  = idk & (FILTERS - 1);
            float dk = dt[e];
            float v0 = __expf(-rate[2 * ch + 0] * dk);
            float v1 = __expf(-rate[2 * ch + 1] * dk);
            size_t o = (size_t)(unsigned)segs[e] * SFM + (size_t)(unsigned)idk * 2;
            atomic_fadd_pair(out + o, v0, v1);
        }
    }
}

// ---------------------------------------------------------------------------
// Launch
// Inputs (setup_inputs order): dt[f32,E], times_out[f32,N_OUT] (unused),
// decay_rate[f32,256], successor_kernel_channel_ids[i32,E], segment_ids_out[i32,E]
// Output: float32, n_out * 1024 elements.
// ---------------------------------------------------------------------------
extern "C" void kernel_launch(void* const* d_in, const int* in_sizes, int n_in,
                              void* d_out, int out_size, void* d_ws, size_t ws_size,
                              hipStream_t stream) {
    (void)n_in; (void)d_ws; (void)ws_size;

    const float* dt    = (const float*)d_in[0];
    const float* decay = (const float*)d_in[2];
    const int*   ids   = (const int*)d_in[3];
    const int*   segs  = (const int*)d_in[4];
    float*       out   = (float*)d_out;

    const int n_events = in_sizes[0];

    // 1) zero 256MB output: 2048 blocks x 256 threads, grid-stride B128 stores
    OneHot_zero_kernel<<<2048, 256, 0, stream>>>(out, out_size);

    // 2) scatter: 1024 events per block (4 per thread)
    int eblocks = (n_events + 1023) / 1024;
    if (eblocks < 1) eblocks = 1;
    OneHot_scatter_kernel<<<eblocks, 256, 0, stream>>>(dt, decay, ids, segs, out, n_events);
}